// MySAGE_29386166239371
// MI455X (gfx1250) — compile-verified
//
#include <hip/hip_runtime.h>
#include <hip/hip_bf16.h>

typedef __attribute__((ext_vector_type(2))) float v2f;
typedef __attribute__((ext_vector_type(8))) float v8f;

// ---------------------------------------------------------------------------
// Zero-fill workspace buffers (graph-capture safe; no hipMemset).
// ---------------------------------------------------------------------------
__launch_bounds__(256)
__global__ void sage_zero_kernel(float* __restrict__ p, long n) {
    long i = (long)blockIdx.x * 256 + threadIdx.x;
    if (i < n) p[i] = 0.0f;
}

// ---------------------------------------------------------------------------
// Degree: cnt[d] += 1 for each edge (computed once; graph is static).
// ---------------------------------------------------------------------------
__launch_bounds__(256)
__global__ void sage_degree_kernel(const int* __restrict__ dst,
                                   float* __restrict__ cnt, int E) {
    int e = blockIdx.x * 256 + threadIdx.x;
    if (e < E) atomicAdd(cnt + dst[e], 1.0f);
}

// ---------------------------------------------------------------------------
// Edge scatter: aggr[dst[e]][c] += x[src[e]][c], 64 channels per edge.
// 64 consecutive threads share one edge -> fully coalesced 256B gather and
// 256B atomic burst to one destination row (fp32 atomics resolve in L2;
// entire 25.6MB feature buffer fits in the 192MB L2).
// ---------------------------------------------------------------------------
__launch_bounds__(256)
__global__ void sage_scatter_kernel(const float* __restrict__ xin,
                                    const int* __restrict__ src,
                                    const int* __restrict__ dst,
                                    float* __restrict__ aggr, int E) {
    long tid = (long)blockIdx.x * 256 + threadIdx.x;
    int  e   = (int)(tid >> 6);
    int  c   = (int)(tid & 63);
    if (e < E) {
        int s = src[e], d = dst[e];
        atomicAdd(aggr + (long)d * 64 + c, xin[(long)s * 64 + c]);
    }
}

// ---------------------------------------------------------------------------
// Combine: out = [ReLU]( (aggr/deg) @ Wl + bl + x @ Wr ), INC = 64.
// One wave32 owns a 16-node tile; fp32 WMMA 16x16x4 over K=64 (16 steps),
// two GEMMs accumulated into the same C fragment. Weights staged in LDS.
//
// Fragment layouts per CDNA5 ISA 7.12.2 (wave32):
//   A 16x4 f32 : lane&15 = M ; VGPR{0,1} = K {kb, kb+1}, kb = (lane>>4)*2
//   B 4x16 f32 : lane&15 = N ; VGPR{0,1} = K {kb, kb+1}
//   C 16x16 f32: lane&15 = N ; VGPR r = row r (lanes 0-15) / r+8 (lanes 16-31)
// ---------------------------------------------------------------------------
template <int OUTC, bool RELU>
__launch_bounds__(256)
__global__ void sage_combine_kernel(const float* __restrict__ xin,   // [N,64]
                                    const float* __restrict__ aggr,  // [N,64]
                                    const float* __restrict__ cnt,   // [N]
                                    const float* __restrict__ Wl,    // [64,OUTC]
                                    const float* __restrict__ bl,    // [OUTC]
                                    const float* __restrict__ Wr,    // [64,OUTC]
                                    float* __restrict__ out,         // [N,OUTC]
                                    int N) {
    __shared__ float sWl[64 * OUTC];
    __shared__ float sWr[64 * OUTC];
    __shared__ float sb[OUTC];

    const int tid = threadIdx.x;
    for (int i = tid; i < 64 * OUTC; i += 256) {
        sWl[i] = Wl[i];
        sWr[i] = Wr[i];
    }
    if (tid < OUTC) sb[tid] = bl[tid];
    __syncthreads();

    const int lane = tid & 31;
    const int wave = tid >> 5;
    const int base = (blockIdx.x * 8 + wave) * 16;   // 8 waves -> 128 nodes/block
    if (base >= N) return;                            // wave-uniform exit

    const int row = base + (lane & 15);
    const int rc  = row < N ? row : (N - 1);          // clamp, no divergence
    const int hi  = lane >> 4;                        // 0 | 1 (half-wave)
    const int col = lane & 15;

    const float deg = cnt[rc];
    const float inv = 1.0f / (deg > 1.0f ? deg : 1.0f);

    // Preload A fragments for both GEMMs (reused across all OUTC/16 tiles).
    v2f afrag[16], xfrag[16];
    const float* arow = aggr + (long)rc * 64;
    const float* xrow = xin  + (long)rc * 64;
#pragma unroll
    for (int k4 = 0; k4 < 16; ++k4) {
        const int kk = k4 * 4 + hi * 2;
        v2f a; a.x = arow[kk] * inv; a.y = arow[kk + 1] * inv;
        afrag[k4] = a;
        v2f xv; xv.x = xrow[kk]; xv.y = xrow[kk + 1];
        xfrag[k4] = xv;
    }

#pragma unroll
    for (int j = 0; j < OUTC / 16; ++j) {
        const float bias = sb[j * 16 + col];
        v8f acc;
#pragma unroll
        for (int r = 0; r < 8; ++r) acc[r] = bias;

        // (aggr/deg) @ Wl
#pragma unroll
        for (int k4 = 0; k4 < 16; ++k4) {
            const int kk = k4 * 4 + hi * 2;
            v2f b;
            b.x = sWl[kk * OUTC + j * 16 + col];
            b.y = sWl[(kk + 1) * OUTC + j * 16 + col];
            acc = __builtin_amdgcn_wmma_f32_16x16x4_f32(
                false, afrag[k4], false, b, (short)0, acc, false, false);
        }
        // x @ Wr
#pragma unroll
        for (int k4 = 0; k4 < 16; ++k4) {
            const int kk = k4 * 4 + hi * 2;
            v2f b;
            b.x = sWr[kk * OUTC + j * 16 + col];
            b.y = sWr[(kk + 1) * OUTC + j * 16 + col];
            acc = __builtin_amdgcn_wmma_f32_16x16x4_f32(
                false, xfrag[k4], false, b, (short)0, acc, false, false);
        }

        if (RELU) {
#pragma unroll
            for (int r = 0; r < 8; ++r) acc[r] = acc[r] > 0.0f ? acc[r] : 0.0f;
        }

#pragma unroll
        for (int r = 0; r < 8; ++r) {
            const int m    = r + hi * 8;
            const int grow = base + m;
            if (grow < N) out[(long)grow * OUTC + j * 16 + col] = acc[r];
        }
    }
}

// ---------------------------------------------------------------------------
// Launch: degree once, then 3x (zero aggr -> scatter -> WMMA combine).
// Workspace: cnt[N] | aggr[N*64] | h1[N*64] | h2[N*64]  (~77 MB, fits L2).
// ---------------------------------------------------------------------------
extern "C" void kernel_launch(void* const* d_in, const int* in_sizes, int n_in,
                              void* d_out, int out_size, void* d_ws, size_t ws_size,
                              hipStream_t stream) {
    const int N = in_sizes[0] / 64;       // 100000
    const int E = in_sizes[1] / 2;        // 1600000

    const float* x   = (const float*)d_in[0];
    const int*   ei  = (const int*)d_in[1];
    const int*   src = ei;                // edge_index[0]
    const int*   dst = ei + E;            // edge_index[1]
    const float* W1l = (const float*)d_in[2];
    const float* b1l = (const float*)d_in[3];
    const float* W1r = (const float*)d_in[4];
    const float* W2l = (const float*)d_in[5];
    const float* b2l = (const float*)d_in[6];
    const float* W2r = (const float*)d_in[7];
    const float* W3l = (const float*)d_in[8];
    const float* b3l = (const float*)d_in[9];
    const float* W3r = (const float*)d_in[10];

    float* ws   = (float*)d_ws;
    float* cnt  = ws;
    float* aggr = ws + N;
    float* h1   = aggr + (size_t)N * 64;
    float* h2   = h1 + (size_t)N * 64;
    float* outp = (float*)d_out;

    const dim3 blk(256);
    auto gridFor = [](long n) { return dim3((unsigned)((n + 255) / 256)); };
    const long agN   = (long)N * 64;
    const dim3 cgrid((N + 127) / 128);

    // Degree (shared by all three layers).
    sage_zero_kernel<<<gridFor(N), blk, 0, stream>>>(cnt, N);
    sage_degree_kernel<<<gridFor(E), blk, 0, stream>>>(dst, cnt, E);

    // Layer 1: 64 -> 64, ReLU
    sage_zero_kernel<<<gridFor(agN), blk, 0, stream>>>(aggr, agN);
    sage_scatter_kernel<<<gridFor((long)E * 64), blk, 0, stream>>>(x, src, dst, aggr, E);
    sage_combine_kernel<64, true><<<cgrid, blk, 0, stream>>>(x, aggr, cnt, W1l, b1l, W1r, h1, N);

    // Layer 2: 64 -> 64, ReLU
    sage_zero_kernel<<<gridFor(agN), blk, 0, stream>>>(aggr, agN);
    sage_scatter_kernel<<<gridFor((long)E * 64), blk, 0, stream>>>(h1, src, dst, aggr, E);
    sage_combine_kernel<64, true><<<cgrid, blk, 0, stream>>>(h1, aggr, cnt, W2l, b2l, W2r, h2, N);

    // Layer 3: 64 -> 32, no ReLU, straight to d_out
    sage_zero_kernel<<<gridFor(agN), blk, 0, stream>>>(aggr, agN);
    sage_scatter_kernel<<<gridFor((long)E * 64), blk, 0, stream>>>(h2, src, dst, aggr, E);
    sage_combine_kernel<32, false><<<cgrid, blk, 0, stream>>>(h2, aggr, cnt, W3l, b3l, W3r, outp, N);
}